// NoiseRouter_71141838291439
// MI455X (gfx1250) — compile-verified
//
#include <hip/hip_runtime.h>
#include <math.h>

typedef float v2f __attribute__((ext_vector_type(2)));
typedef float v8f __attribute__((ext_vector_type(8)));

#define B_    4
#define S_    4096
#define D_    2048
#define E_    8
#define K_    2
#define NTOK  (B_ * S_)     // 16384 tokens
#define NPAIR (D_ / 2)      // 1024 k-pairs

// ---------------------------------------------------------------------------
// Prep: pack W_gate|W_noise into a [NPAIR][16] float2 array in d_ws.
// wc[p][col] = { Wsel[2p][col'], Wsel[2p+1][col'] }, col<8 -> gate, else noise.
// Makes the main kernel's B-matrix load one coalesced 256B burst per wave.
// ---------------------------------------------------------------------------
__global__ __launch_bounds__(256) void noise_router_prep(
    const float* __restrict__ Wg, const float* __restrict__ Wn,
    float2* __restrict__ wc)
{
    int i = blockIdx.x * blockDim.x + threadIdx.x;   // 0 .. NPAIR*16-1
    if (i >= NPAIR * 16) return;
    int p   = i >> 4;
    int col = i & 15;
    const float* src = (col < 8) ? (Wg + col) : (Wn + (col - 8));
    float lo = src[(size_t)(2 * p + 0) * E_];
    float hi = src[(size_t)(2 * p + 1) * E_];
    wc[i] = make_float2(lo, hi);
}

// ---------------------------------------------------------------------------
// Main: per wave, 16 tokens. Fused dual-GEMM via one f32 WMMA chain
// (B = [W_gate | W_noise], 16 cols), then top-2 + masked softmax.
// ---------------------------------------------------------------------------
__global__ __launch_bounds__(256) void noise_router_main(
    const float* __restrict__ x,      // [NTOK, D]
    const float* __restrict__ noise,  // [NTOK, E]
    const float* __restrict__ bg,     // [E]
    const float* __restrict__ bn,     // [E]
    const float2* __restrict__ wc,    // [NPAIR, 16] packed weights
    float* __restrict__ outScores,    // [NTOK, E]
    float* __restrict__ outIds)       // [NTOK, K] (ids written as floats)
{
    __shared__ float ldsT[8][16][16]; // [wave][token-row][col]  (8 KB)

    const int lane    = threadIdx.x & 31;
    const int wave    = threadIdx.x >> 5;
    const int tokBase = blockIdx.x * 128 + wave * 16;
    const int half    = lane >> 4;    // 0 or 1
    const int l15     = lane & 15;

    // A-matrix layout (16x4 f32): lane = half*16 + row; VGPR0/1 = K half*2+{0,1}
    const float*  xrow = x  + (size_t)(tokBase + l15) * D_ + half * 2;
    const float2* wcl  = wc + half * 16 + l15;   // B: pair p = k0/2 + half, col = l15

    v8f c0 = {};
    v8f c1 = {};
#pragma unroll 4
    for (int k0 = 0; k0 < D_; k0 += 8) {
        v2f a0 = *(const v2f*)(xrow + k0);
        float2 w0 = wcl[(k0 >> 1) * 16];
        v2f b0; b0[0] = w0.x; b0[1] = w0.y;
        c0 = __builtin_amdgcn_wmma_f32_16x16x4_f32(
                 false, a0, false, b0, (short)0, c0, false, false);

        v2f a1 = *(const v2f*)(xrow + k0 + 4);
        float2 w1 = wcl[((k0 + 4) >> 1) * 16];
        v2f b1; b1[0] = w1.x; b1[1] = w1.y;
        c1 = __builtin_amdgcn_wmma_f32_16x16x4_f32(
                 false, a1, false, b1, (short)0, c1, false, false);
    }

    // C/D layout: lane L, VGPR v -> row = v + 8*half, col = l15.
    // Dump the whole 16x16 tile to LDS (no shuffles, no exec masking);
    // cols 0-7 = gate(e), cols 8-15 = noise_gate(e).
#pragma unroll
    for (int v = 0; v < 8; ++v) {
        ldsT[wave][v + half * 8][l15] = c0[v] + c1[v];
    }
    __syncthreads();

    if (lane < 16) {
        const int tok = tokBase + lane;
        // Row is 64B and 16B-aligned: read as 4x b128.
        const float4* rowp = (const float4*)&ldsT[wave][lane][0];
        float4 g0 = rowp[0], g1 = rowp[1];   // gate       e = 0..7
        float4 h0 = rowp[2], h1 = rowp[3];   // noise-gate e = 0..7
        const float4* np = (const float4*)(noise + (size_t)tok * E_);
        float4 z0 = np[0], z1 = np[1];
        const float4* bgp = (const float4*)bg;
        const float4* bnp = (const float4*)bn;
        float4 bg0 = bgp[0], bg1 = bgp[1];
        float4 bn0 = bnp[0], bn1 = bnp[1];

        float l[8];
        l[0] = g0.x + h0.x + z0.x + bg0.x + bn0.x;
        l[1] = g0.y + h0.y + z0.y + bg0.y + bn0.y;
        l[2] = g0.z + h0.z + z0.z + bg0.z + bn0.z;
        l[3] = g0.w + h0.w + z0.w + bg0.w + bn0.w;
        l[4] = g1.x + h1.x + z1.x + bg1.x + bn1.x;
        l[5] = g1.y + h1.y + z1.y + bg1.y + bn1.y;
        l[6] = g1.z + h1.z + z1.z + bg1.z + bn1.z;
        l[7] = g1.w + h1.w + z1.w + bg1.w + bn1.w;

        // top-1 (strict >, lowest index wins ties — matches jax.lax.top_k)
        int i0 = 0; float v0 = l[0];
#pragma unroll
        for (int e = 1; e < 8; ++e) if (l[e] > v0) { v0 = l[e]; i0 = e; }
        // top-2 excluding i0
        int i1 = -1; float v1 = -INFINITY;
#pragma unroll
        for (int e = 0; e < 8; ++e)
            if (e != i0 && l[e] > v1) { v1 = l[e]; i1 = e; }

        // softmax over {v0, v1} (others are -inf -> 0); v0 is the max
        float e1  = expf(v1 - v0);
        float inv = 1.0f / (1.0f + e1);
        float s0  = inv;
        float s1  = e1 * inv;

        float sc[8];
#pragma unroll
        for (int e = 0; e < 8; ++e)
            sc[e] = (e == i0) ? s0 : ((e == i1) ? s1 : 0.0f);

        float4* outp = (float4*)(outScores + (size_t)tok * E_);  // 32B-aligned
        outp[0] = make_float4(sc[0], sc[1], sc[2], sc[3]);
        outp[1] = make_float4(sc[4], sc[5], sc[6], sc[7]);
        *(float2*)(outIds + (size_t)tok * K_) =
            make_float2((float)i0, (float)i1);
    }
}

// ---------------------------------------------------------------------------
extern "C" void kernel_launch(void* const* d_in, const int* in_sizes, int n_in,
                              void* d_out, int out_size, void* d_ws, size_t ws_size,
                              hipStream_t stream)
{
    const float* x     = (const float*)d_in[0];
    const float* noise = (const float*)d_in[1];
    const float* Wg    = (const float*)d_in[2];
    const float* bg    = (const float*)d_in[3];
    const float* Wn    = (const float*)d_in[4];
    const float* bn    = (const float*)d_in[5];

    float* out       = (float*)d_out;
    float* outScores = out;                        // [NTOK*E] floats
    float* outIds    = out + (size_t)NTOK * E_;    // [NTOK*K] ids-as-floats

    float2* wc = (float2*)d_ws;                    // 128 KB packed weights

    noise_router_prep<<<(NPAIR * 16 + 255) / 256, 256, 0, stream>>>(Wg, Wn, wc);
    noise_router_main<<<NTOK / 128, 256, 0, stream>>>(
        x, noise, bg, bn, wc, outScores, outIds);
}